// RNNDecoder_14834817040711
// MI455X (gfx1250) — compile-verified
//
#include <hip/hip_runtime.h>
#include <math.h>

typedef float v2f __attribute__((ext_vector_type(2)));
typedef float v8f __attribute__((ext_vector_type(8)));

#define B_      16
#define ZDIM_   128
#define EDIM_   512
#define HDIM_   512
#define V_      50257
#define STEPS_  128
#define NWG     128
#define NTHR    256
#define WPW     (NTHR / 32)          // waves per workgroup (wave32)
#define TOTW    (NWG * WPW)          // total waves in grid
#define NT_LOG  ((V_ + 15) / 16)     // 3142 logit tiles of 16 cols
#define NT_GH   ((3 * HDIM_) / 16)   // 96 gh tiles
#define CHUNKW  ((V_ + NWG - 1) / NWG)

struct WS {
    unsigned long long argBest;      // packed (ordered-float<<32)|(~idx)
    unsigned barCount;
    unsigned barSense;
    unsigned rowMaxBits[16];         // ordered-uint encoded row maxima
    float    rowLogSum[16];
    float    partial[NWG * 16];      // per-WG per-row exp-sum partials
    float    ix[3 * HDIM_];
    float    hbuf[2][B_ * HDIM_];    // ping-pong hidden state
    float    gh[B_ * 3 * HDIM_];
};

__device__ __forceinline__ unsigned fenc(float f) {
    unsigned u = __float_as_uint(f);
    return (u & 0x80000000u) ? ~u : (u | 0x80000000u);
}
__device__ __forceinline__ float fdec(unsigned e) {
    unsigned u = (e & 0x80000000u) ? (e ^ 0x80000000u) : ~e;
    return __uint_as_float(u);
}

__device__ __forceinline__ void gbar(WS* w) {
    __threadfence();
    __syncthreads();
    if (threadIdx.x == 0) {
        volatile unsigned* vs = &w->barSense;
        unsigned s = *vs;
        unsigned arrived = atomicAdd(&w->barCount, 1u);
        if (arrived == NWG - 1) {
            w->barCount = 0;
            __threadfence();
            atomicAdd(&w->barSense, 1u);
        } else {
            while (*vs == s) { __builtin_amdgcn_s_sleep(1); }
        }
    }
    __syncthreads();
}

__global__ void __launch_bounds__(NTHR)
rnn_decoder_kernel(const float* __restrict__ z, const float* __restrict__ embed,
                   const float* __restrict__ decode_w, const float* __restrict__ decode_b,
                   const float* __restrict__ w_ih, const float* __restrict__ w_hh,
                   const float* __restrict__ b_ih, const float* __restrict__ b_hh,
                   const float* __restrict__ out_w, const float* __restrict__ out_b,
                   float* __restrict__ out, WS* __restrict__ ws)
{
    const int tid   = threadIdx.x;
    const int lane  = tid & 31;
    const int gwave = blockIdx.x * WPW + (tid >> 5);
    const int gthr  = blockIdx.x * NTHR + tid;
    const int m     = lane & 15;            // A row / B col-in-tile
    const int ko    = (lane < 16) ? 0 : 2;  // K sub-offset (ISA 16x4 fp32 A/B layout)
    const int mhi   = (lane < 16) ? 0 : 8;  // C/D row offset per ISA layout

    __shared__ float sH[B_ * HDIM_];        // h_new staged per-WG for logits GEMM
    __shared__ float sred[NTHR];
    __shared__ unsigned long long sbest[NTHR];

    // ---------------- init: h0 = z @ decode_w.T + decode_b ; ix = embed[EOS] @ w_ih.T + b_ih
    for (int i = gthr; i < B_ * HDIM_; i += NWG * NTHR) {
        int b = i / HDIM_, j = i - b * HDIM_;
        const float* zr = z + b * ZDIM_;
        const float* wr = decode_w + (size_t)j * ZDIM_;
        float acc = decode_b[j];
        for (int k = 0; k < ZDIM_; ++k) acc = fmaf(zr[k], wr[k], acc);
        ws->hbuf[0][i] = acc;
    }
    for (int i = gthr; i < 3 * HDIM_; i += NWG * NTHR) {
        const float* xr = embed + 2 * EDIM_;            // EOS_ID = 2
        const float* wr = w_ih + (size_t)i * EDIM_;
        float acc = b_ih[i];
        for (int k = 0; k < EDIM_; ++k) acc = fmaf(xr[k], wr[k], acc);
        ws->ix[i] = acc;
    }
    gbar(ws);

    int cur = 0;
    for (int step = 0; step < STEPS_; ++step) {
        const int nxt = cur ^ 1;
        const float* hp = ws->hbuf[cur];
        float* hn = ws->hbuf[nxt];
        float* logits = out + (size_t)step * B_ * V_;

        // -------- Phase A1: gh = h @ w_hh.T + b_hh  (96 WMMA tiles, fp32 16x16x4)
        for (int t = gwave; t < NT_GH; t += TOTW) {
            const int n = t * 16 + m;
            const float* wrow = w_hh + (size_t)n * HDIM_;
            const float* arow = hp + (size_t)m * HDIM_;
            {   // prefetch next tile's weight row into WGP$/L2
                int tn = t + TOTW;
                if (tn < NT_GH)
                    __builtin_prefetch(w_hh + (size_t)(tn * 16 + m) * HDIM_, 0, 3);
            }
            v8f c = {};
            #pragma unroll 8
            for (int k = 0; k < HDIM_; k += 4) {
                v2f a = { arow[k + ko], arow[k + ko + 1] };
                v2f b = { wrow[k + ko], wrow[k + ko + 1] };
                c = __builtin_amdgcn_wmma_f32_16x16x4_f32(false, a, false, b,
                                                          (short)0, c, false, false);
            }
            const float bias = b_hh[n];
            #pragma unroll
            for (int v = 0; v < 8; ++v)
                ws->gh[(size_t)(v + mhi) * (3 * HDIM_) + n] = c[v] + bias;
        }
        gbar(ws);

        // -------- Phase A2: GRU gates -> h_new
        for (int i = gthr; i < B_ * HDIM_; i += NWG * NTHR) {
            int b = i / HDIM_, j = i - b * HDIM_;
            const float* g = ws->gh + (size_t)b * (3 * HDIM_);
            float r  = 1.f / (1.f + expf(-(ws->ix[j] + g[j])));
            float zg = 1.f / (1.f + expf(-(ws->ix[HDIM_ + j] + g[HDIM_ + j])));
            float n  = tanhf(ws->ix[2 * HDIM_ + j] + r * g[2 * HDIM_ + j]);
            hn[i] = (1.f - zg) * n + zg * hp[i];
        }
        gbar(ws);

        // -------- Phase B: logits = h_new @ out_w.T + out_b  (WMMA, A from LDS)
        for (int i = tid; i < B_ * HDIM_; i += NTHR) sH[i] = hn[i];
        __syncthreads();
        {
            float vmax[8];
            #pragma unroll
            for (int v = 0; v < 8; ++v) vmax[v] = -3.4e38f;
            const float* arow = sH + (size_t)m * HDIM_;
            for (int t = gwave; t < NT_LOG; t += TOTW) {
                const int n = t * 16 + m;
                const int nn = (n < V_) ? n : (V_ - 1);
                const float* wrow = out_w + (size_t)nn * HDIM_;
                {   // prefetch next tile's out_w row (global_prefetch_b8)
                    int tn = t + TOTW;
                    if (tn < NT_LOG) {
                        int np = tn * 16 + m;
                        if (np >= V_) np = V_ - 1;
                        __builtin_prefetch(out_w + (size_t)np * HDIM_, 0, 3);
                    }
                }
                v8f c = {};
                #pragma unroll 8
                for (int k = 0; k < HDIM_; k += 4) {
                    v2f a = { arow[k + ko], arow[k + ko + 1] };
                    v2f b = { wrow[k + ko], wrow[k + ko + 1] };
                    c = __builtin_amdgcn_wmma_f32_16x16x4_f32(false, a, false, b,
                                                              (short)0, c, false, false);
                }
                const float bias = (n < V_) ? out_b[n] : 0.f;
                #pragma unroll
                for (int v = 0; v < 8; ++v) {
                    float val = c[v] + bias;
                    if (n < V_) {
                        logits[(size_t)(v + mhi) * V_ + n] = val;
                        vmax[v] = fmaxf(vmax[v], val);
                    }
                }
            }
            // per-row max: reduce across the 16-lane half-wave, then one atomic per row
            #pragma unroll
            for (int v = 0; v < 8; ++v) {
                float x = vmax[v];
                x = fmaxf(x, __shfl_xor(x, 1, 32));
                x = fmaxf(x, __shfl_xor(x, 2, 32));
                x = fmaxf(x, __shfl_xor(x, 4, 32));
                x = fmaxf(x, __shfl_xor(x, 8, 32));
                if ((lane & 15) == 0)
                    atomicMax(&ws->rowMaxBits[v + mhi], fenc(x));
            }
        }
        gbar(ws);

        // -------- Phase C: deterministic per-WG exp-sum partials
        {
            const int r = tid >> 4, j = tid & 15;
            const float rmax = fdec(ws->rowMaxBits[r]);
            const int c0 = blockIdx.x * CHUNKW;
            float acc = 0.f;
            for (int c = c0 + j; c < c0 + CHUNKW; c += 16)
                if (c < V_) acc += expf(logits[(size_t)r * V_ + c] - rmax);
            sred[tid] = acc;
            __syncthreads();
            if (j == 0) {
                float s = 0.f;
                #pragma unroll
                for (int q = 0; q < 16; ++q) s += sred[(r << 4) + q];
                ws->partial[blockIdx.x * 16 + r] = s;
            }
        }
        gbar(ws);

        // -------- Phase D: reduce partials -> log-sum (single WG, fixed order)
        if (blockIdx.x == 0 && tid < 16) {
            float s = 0.f;
            for (int q = 0; q < NWG; ++q) s += ws->partial[q * 16 + tid];
            ws->rowLogSum[tid] = logf(s);
        }
        gbar(ws);

        // -------- Phase E: logp in place + global argmax (packed u64 max-atomic)
        {
            float rowOff[16];
            #pragma unroll
            for (int r = 0; r < 16; ++r)
                rowOff[r] = fdec(ws->rowMaxBits[r]) + ws->rowLogSum[r];
            unsigned long long best = 0;
            for (int i = gthr; i < B_ * V_; i += NWG * NTHR) {
                int r = i / V_;
                float lv = logits[i] - rowOff[r];
                logits[i] = lv;
                unsigned long long p = ((unsigned long long)fenc(lv) << 32)
                                     | (unsigned long long)(0xFFFFFFFFu - (unsigned)i);
                if (p > best) best = p;
            }
            sbest[tid] = best;
            __syncthreads();
            for (int off = NTHR / 2; off > 0; off >>= 1) {
                if (tid < off && sbest[tid + off] > sbest[tid]) sbest[tid] = sbest[tid + off];
                __syncthreads();
            }
            if (tid == 0) atomicMax(&ws->argBest, sbest[0]);
        }
        gbar(ws);

        // -------- Phase F: emit token, reset per-step accumulators (WG0)
        if (blockIdx.x == 0 && tid == 0) {
            unsigned idx = 0xFFFFFFFFu - (unsigned)(ws->argBest & 0xFFFFFFFFull);
            ((int*)out)[(size_t)STEPS_ * B_ * V_ + step] = (int)idx;
            ws->argBest = 0;
            #pragma unroll
            for (int q = 0; q < 16; ++q) ws->rowMaxBits[q] = 0;
        }
        cur = nxt;
    }
}

extern "C" void kernel_launch(void* const* d_in, const int* in_sizes, int n_in,
                              void* d_out, int out_size, void* d_ws, size_t ws_size,
                              hipStream_t stream) {
    const float* z        = (const float*)d_in[0];
    const float* embed    = (const float*)d_in[1];
    const float* decode_w = (const float*)d_in[2];
    const float* decode_b = (const float*)d_in[3];
    const float* w_ih     = (const float*)d_in[4];
    const float* w_hh     = (const float*)d_in[5];
    const float* b_ih     = (const float*)d_in[6];
    const float* b_hh     = (const float*)d_in[7];
    const float* out_w    = (const float*)d_in[8];
    const float* out_b    = (const float*)d_in[9];
    (void)in_sizes; (void)n_in; (void)out_size; (void)ws_size;

    WS* ws = (WS*)d_ws;
    hipMemsetAsync(d_ws, 0, sizeof(WS), stream);   // graph-capture-legal init of barriers/accums
    rnn_decoder_kernel<<<NWG, NTHR, 0, stream>>>(z, embed, decode_w, decode_b,
                                                 w_ih, w_hh, b_ih, b_hh,
                                                 out_w, out_b, (float*)d_out, ws);
}